// DistanceTransform_17136919511192
// MI455X (gfx1250) — compile-verified
//
#include <hip/hip_runtime.h>
#include <math.h>

// Problem constants (reference: B=16, H=512, W=512, float32)
#define EDT_B 16
#define EDT_H 512
#define EDT_W 512
#define EDT_HW (EDT_H * EDT_W)
#define EDT_L 512              // line length for the min-plus passes
#define EDT_TILE 8             // lines per block
#define EDT_PAD (EDT_TILE + 1) // LDS row stride: gcd(9,64)=1 -> conflict-free
#define EDT_THREADS 256
#define EDT_INF 1.0e9f         // matches reference INF

// CDNA5 async global->LDS path (confirmed present on this toolchain; the
// round-1 diagnostic showed arg0 is an AS1 int*). Guarded anyway.
#if defined(__has_builtin)
#if __has_builtin(__builtin_amdgcn_global_load_async_to_lds_b32) && \
    __has_builtin(__builtin_amdgcn_s_wait_asynccnt)
#define EDT_HAVE_ASYNC_LDS 1
#endif
#endif

typedef __attribute__((address_space(1))) int* edt_gptr_t;
typedef __attribute__((address_space(3))) int* edt_lptr_t;

// ---------------------------------------------------------------------------
// Pass 0: exact 1-D distance-to-nearest-zero along axis B (length 16),
// squared. One thread per (h,w) column; scans live entirely in registers.
// Bitwise-matches the reference: c = min(c+1, d0) with c0 = 1e9 (1e9+1
// rounds back to 1e9 in fp32, same as jnp).
// ---------------------------------------------------------------------------
__global__ __launch_bounds__(EDT_THREADS)
void edt_pass0_axisB(const float* __restrict__ x, float* __restrict__ d2)
{
    int idx = blockIdx.x * blockDim.x + threadIdx.x;
    if (idx >= EDT_HW) return;
    __builtin_prefetch(x + idx, 0, 1);  // global_prefetch_b8 on gfx1250

    float d0s[EDT_B];
    float fwd[EDT_B];
    float c = EDT_INF;
#pragma unroll
    for (int b = 0; b < EDT_B; ++b) {
        float v = x[b * EDT_HW + idx];
        float d0 = (v == 0.0f) ? 0.0f : EDT_INF;  // NaN -> foreground (false)
        d0s[b] = d0;
        c = fminf(c + 1.0f, d0);
        fwd[b] = c;
    }
    c = EDT_INF;
#pragma unroll
    for (int b = EDT_B - 1; b >= 0; --b) {
        c = fminf(c + 1.0f, d0s[b]);
        float d = fminf(fwd[b], c);
        d2[b * EDT_HW + idx] = d * d;
    }
}

// ---------------------------------------------------------------------------
// Passes 1 & 2: exact min-plus parabola along one axis.
//   out[i] = min_k ( (i-k)^2 + g[k] ),  L = 512.
// Each block owns one image b and a tile of EDT_TILE lines. The tile is
// staged into LDS transposed with pad stride EDT_PAD (bank-conflict-free for
// both load layouts and for the lane-indexed reduce reads). Each thread
// keeps 16 accumulators in VGPRs; one LDS read per k feeds 16 fma+min.
// In-place safe: every tile's global reads/writes are block-exclusive and
// separated from the LDS load by a barrier.
// (i-k)^2 <= 511^2 < 2^24 is exact in fp32, so fmaf(t,t,g) rounds
// identically to the reference's t*t then +g.
// ---------------------------------------------------------------------------
__global__ __launch_bounds__(EDT_THREADS)
void edt_minplus(const float* __restrict__ in, float* __restrict__ out,
                 const float* __restrict__ xsrc,
                 int line_stride, int k_stride, int finalize)
{
    __shared__ float sg[EDT_L * EDT_PAD];  // 512*9*4 = 18 KB

    const int tiles_per_img = EDT_L / EDT_TILE;  // 64
    const int b     = blockIdx.x / tiles_per_img;
    const int line0 = (blockIdx.x % tiles_per_img) * EDT_TILE;
    const int base  = b * EDT_HW;
    const int tid   = threadIdx.x;

    // --- stage tile into LDS (coalesced decomposition per pass layout) ---
    for (int idx = tid; idx < EDT_L * EDT_TILE; idx += EDT_THREADS) {
        int k, line;
        if (k_stride == 1) {            // W-pass: k contiguous in memory
            line = idx >> 9;            // idx / 512
            k    = idx & (EDT_L - 1);
        } else {                        // H-pass: line contiguous in memory
            k    = idx >> 3;            // idx / 8
            line = idx & (EDT_TILE - 1);
        }
        int ga = base + (line0 + line) * line_stride + k * k_stride;
        int la = k * EDT_PAD + line;
#ifdef EDT_HAVE_ASYNC_LDS
        float* src_nc = const_cast<float*>(in + ga);
        __builtin_amdgcn_global_load_async_to_lds_b32(
            (edt_gptr_t)src_nc, (edt_lptr_t)(&sg[la]), 0, 0);
#else
        sg[la] = in[ga];
#endif
    }
#ifdef EDT_HAVE_ASYNC_LDS
    __builtin_amdgcn_s_wait_asynccnt(0);
#endif
    __syncthreads();

    // --- reduce: thread -> (line = tid%8, i in [i0, i0+16)) ---
    const int line = tid & (EDT_TILE - 1);
    const int i0   = (tid >> 3) * 16;   // 32 groups * 16 = 512 outputs/line

    float acc[16];
#pragma unroll
    for (int j = 0; j < 16; ++j) acc[j] = 3.0e38f;

    for (int k = 0; k < EDT_L; ++k) {
        float g = sg[k * EDT_PAD + line];   // broadcast within 8-lane groups
        float d = (float)(i0 - k);
#pragma unroll
        for (int j = 0; j < 16; ++j) {
            float t = d + (float)j;
            acc[j] = fminf(acc[j], fmaf(t, t, g));
        }
    }

    // --- write back (in-place safe; optional finalize fuses sqrt + NaN) ---
#pragma unroll
    for (int j = 0; j < 16; ++j) {
        int i  = i0 + j;
        int ga = base + (line0 + line) * line_stride + i * k_stride;
        if (finalize) {
            float xv = xsrc[ga];
            float r  = sqrtf(acc[j]);
            out[ga] = (xv != xv) ? __builtin_nanf("") : r;
        } else {
            out[ga] = acc[j];
        }
    }
}

// ---------------------------------------------------------------------------
// Launch: pass0 (x -> out as d^2), min-plus along H in-place, min-plus along
// W in-place with fused sqrt + NaN mask. No workspace needed.
// ---------------------------------------------------------------------------
extern "C" void kernel_launch(void* const* d_in, const int* in_sizes, int n_in,
                              void* d_out, int out_size, void* d_ws, size_t ws_size,
                              hipStream_t stream)
{
    (void)in_sizes; (void)n_in; (void)out_size; (void)d_ws; (void)ws_size;
    const float* x  = (const float*)d_in[0];
    float*      out = (float*)d_out;

    dim3 blk(EDT_THREADS);
    edt_pass0_axisB<<<dim3(EDT_HW / EDT_THREADS), blk, 0, stream>>>(x, out);

    const int nblocks = EDT_B * (EDT_L / EDT_TILE);  // 16 * 64 = 1024
    // Axis H: k runs along H (stride W), lines are w (stride 1)
    edt_minplus<<<dim3(nblocks), blk, 0, stream>>>(out, out, nullptr,
                                                   /*line_stride=*/1,
                                                   /*k_stride=*/EDT_W,
                                                   /*finalize=*/0);
    // Axis W: k runs along W (stride 1), lines are h (stride W); finalize
    edt_minplus<<<dim3(nblocks), blk, 0, stream>>>(out, out, x,
                                                   /*line_stride=*/EDT_W,
                                                   /*k_stride=*/1,
                                                   /*finalize=*/1);
}